// GATLayer_61332132986980
// MI455X (gfx1250) — compile-verified
//
#include <hip/hip_runtime.h>
#include <hip/hip_bf16.h>

typedef float v2f __attribute__((ext_vector_type(2)));
typedef float v8f __attribute__((ext_vector_type(8)));

#define IN_FEATS 128
#define HEADS 4
#define OUT_FEATS 32
#define HF 128   // HEADS * OUT_FEATS

// ---------------------------------------------------------------------------
// Kernel 0: zero the output accumulator (edge kernel uses atomics into it).
// ---------------------------------------------------------------------------
__global__ void gat_zero_f32(float* __restrict__ p, int n) {
    int i = blockIdx.x * blockDim.x + threadIdx.x;
    if (i < n) p[i] = 0.0f;
}

// ---------------------------------------------------------------------------
// Kernel 1: h = x @ W + bias  via V_WMMA_F32_16X16X4_F32.
// Block = 256 threads (8 waves). Wave w computes the 16x16 tile
// (rows blockIdx.x*16 .. +15, cols w*16 .. +15). K loop of 32 WMMAs.
//
// Fragment layouts (CDNA5 ISA 7.12.2, 32-bit, 16x16x4):
//   A (16x4):  lane holds row M=lane&15; vgpr j holds K = k + 2*(lane>>4) + j
//   B (4x16):  lane holds col N=lane&15; vgpr j holds K = k + 2*(lane>>4) + j
//   C/D:       vgpr i -> row = i + 8*(lane>>4), col = lane&15
// ---------------------------------------------------------------------------
__global__ __launch_bounds__(256) void gat_gemm_wmma(
    const float* __restrict__ X, const float* __restrict__ W,
    const float* __restrict__ bias, float* __restrict__ H, int nNodes)
{
    const int lane = threadIdx.x & 31;
    const int wave = threadIdx.x >> 5;
    const int half = lane >> 4;       // 0 or 1
    const int l15  = lane & 15;
    const int tm   = blockIdx.x * 16;
    const int tn   = wave * 16;       // 8 waves cover all 128 output cols

    int rowA = tm + l15;
    if (rowA >= nNodes) rowA = nNodes - 1;   // clamp loads (EXEC stays all-1s)
    const int colB = tn + l15;

    v8f c;
    const float b = bias[colB];
#pragma unroll
    for (int i = 0; i < 8; ++i) c[i] = b;

    const float* xrow = X + (size_t)rowA * IN_FEATS;
#pragma unroll
    for (int k = 0; k < IN_FEATS; k += 4) {
        const int kb = k + 2 * half;
        v2f af = *(const v2f*)(xrow + kb);          // A[row][kb], A[row][kb+1]
        v2f bf;
        bf.x = W[(size_t)kb * HF + colB];           // B[kb][col]
        bf.y = W[(size_t)(kb + 1) * HF + colB];     // B[kb+1][col]
        // 8 args: (neg_a, A, neg_b, B, c_mod, C, reuse_a, reuse_b)
        c = __builtin_amdgcn_wmma_f32_16x16x4_f32(
                false, af, false, bf, (short)0, c, false, false);
    }

#pragma unroll
    for (int i = 0; i < 8; ++i) {
        const int row = tm + i + 8 * half;
        if (row < nNodes) H[(size_t)row * HF + colB] = c[i];
    }
}

// ---------------------------------------------------------------------------
// Kernel 2: per-edge attention + scatter. One wave32 per edge.
// Lane l holds float4 = h[., head l>>3, feats 4*(l&7)..+3]  (flat offs 4l..4l+3)
// scores[h][g] = sum_f tanh(hs+hd)[h][f] * a[g][f]:
//   per-lane partial over its 4 feats, xor-reduce over the 8 lanes of the
//   head group, then broadcast each head's 4 scores to every lane.
// ---------------------------------------------------------------------------
__global__ __launch_bounds__(256) void gat_edge(
    const float* __restrict__ H, const float* __restrict__ A,
    const int* __restrict__ src, const int* __restrict__ dst,
    float* __restrict__ out, int nEdges)
{
    const int lane = threadIdx.x & 31;
    const int e = blockIdx.x * 8 + (threadIdx.x >> 5);
    if (e >= nEdges) return;

    const int s = src[e];
    const int d = dst[e];
    const float4 hs = ((const float4*)(H + (size_t)s * HF))[lane];
    const float4 hd = ((const float4*)(H + (size_t)d * HF))[lane];

    const float t0 = tanhf(hs.x + hd.x);
    const float t1 = tanhf(hs.y + hd.y);
    const float t2 = tanhf(hs.z + hd.z);
    const float t3 = tanhf(hs.w + hd.w);

    const int fr = lane & 7;          // which float4 of the 32 feats
    float p[4];
#pragma unroll
    for (int g = 0; g < 4; ++g) {
        const float4 ag = ((const float4*)(A + g * OUT_FEATS))[fr];
        p[g] = t0 * ag.x + t1 * ag.y + t2 * ag.z + t3 * ag.w;
    }
    // reduce over the 8 lanes of this head's group
#pragma unroll
    for (int m = 1; m < 8; m <<= 1) {
#pragma unroll
        for (int g = 0; g < 4; ++g) p[g] += __shfl_xor(p[g], m, 32);
    }
    // gather scores of all 4 heads into every lane (group h base lane = 8h)
    float sc[4][4];
#pragma unroll
    for (int h = 0; h < 4; ++h)
#pragma unroll
        for (int g = 0; g < 4; ++g)
            sc[h][g] = __shfl(p[g], h * 8, 32);

    // elu
#pragma unroll
    for (int h = 0; h < 4; ++h)
#pragma unroll
        for (int g = 0; g < 4; ++g) {
            const float v = sc[h][g];
            sc[h][g] = v > 0.0f ? v : (__expf(v) - 1.0f);
        }

    // softmax over the head axis (per g), w[g] = sum_h att[h][g], folded 1/H
    float w[4];
#pragma unroll
    for (int g = 0; g < 4; ++g) {
        float mx = sc[0][g];
#pragma unroll
        for (int h = 1; h < 4; ++h) mx = fmaxf(mx, sc[h][g]);
        float ex[4], sum = 0.0f;
#pragma unroll
        for (int h = 0; h < 4; ++h) { ex[h] = __expf(sc[h][g] - mx); sum += ex[h]; }
        const float inv = 1.0f / sum;
        float acc = 0.0f;
#pragma unroll
        for (int h = 0; h < 4; ++h) acc += ex[h] * inv;   // matches att.sum(axis=1)
        w[g] = acc * 0.25f;                                // fold mean over heads
    }

    const float wl = w[lane >> 3];
    float* orow = out + (size_t)d * HF + lane * 4;
    __hip_atomic_fetch_add(orow + 0, hs.x * wl, __ATOMIC_RELAXED, __HIP_MEMORY_SCOPE_AGENT);
    __hip_atomic_fetch_add(orow + 1, hs.y * wl, __ATOMIC_RELAXED, __HIP_MEMORY_SCOPE_AGENT);
    __hip_atomic_fetch_add(orow + 2, hs.z * wl, __ATOMIC_RELAXED, __HIP_MEMORY_SCOPE_AGENT);
    __hip_atomic_fetch_add(orow + 3, hs.w * wl, __ATOMIC_RELAXED, __HIP_MEMORY_SCOPE_AGENT);
}

// ---------------------------------------------------------------------------
extern "C" void kernel_launch(void* const* d_in, const int* in_sizes, int n_in,
                              void* d_out, int out_size, void* d_ws, size_t ws_size,
                              hipStream_t stream) {
    const float* x    = (const float*)d_in[0];
    const float* wgt  = (const float*)d_in[1];
    const float* bias = (const float*)d_in[2];
    const float* a    = (const float*)d_in[3];
    const int*   src  = (const int*)d_in[4];
    const int*   dst  = (const int*)d_in[5];
    float* out = (float*)d_out;

    const int nNodes = in_sizes[0] / IN_FEATS;
    const int nEdges = in_sizes[4];
    float* h = (float*)d_ws;   // nNodes*128 floats = 25.6 MB scratch

    // 1) zero accumulator
    gat_zero_f32<<<(out_size + 255) / 256, 256, 0, stream>>>(out, out_size);

    // 2) projection GEMM (WMMA f32 16x16x4)
    const int mtiles = (nNodes + 15) / 16;
    gat_gemm_wmma<<<mtiles, 256, 0, stream>>>(x, wgt, bias, h, nNodes);

    // 3) edge attention + atomic scatter (one wave per edge, 8 edges/block)
    gat_edge<<<(nEdges + 7) / 8, 256, 0, stream>>>(h, a, src, dst, out, nEdges);
}